// GraphConv_5669356834162
// MI455X (gfx1250) — compile-verified
//
#include <hip/hip_runtime.h>

#define N_NODES 50000
#define N_EDGES 800000
#define DIM 128
#define LDS_STRIDE 132   // 128 + 4 pad: conflict-free b64 LDS reads across wave32

typedef __attribute__((ext_vector_type(2))) float v2f;
typedef __attribute__((ext_vector_type(8))) float v8f;

// ---------------------------------------------------------------------------
// Kernel 0: zero degree scratch (2N floats in ws) and the accumulator (d_out)
// ---------------------------------------------------------------------------
__global__ void gc_zero_kernel(float* __restrict__ deg, float* __restrict__ upd) {
    const size_t ndeg = 2u * N_NODES;
    const size_t nupd = (size_t)N_NODES * DIM;
    const size_t total = ndeg + nupd;
    size_t stride = (size_t)gridDim.x * blockDim.x;
    for (size_t i = (size_t)blockIdx.x * blockDim.x + threadIdx.x; i < total; i += stride) {
        if (i < ndeg) deg[i] = 0.0f;
        else          upd[i - ndeg] = 0.0f;
    }
}

// ---------------------------------------------------------------------------
// Kernel 1: degree accumulation (L2-resident float atomics)
// ---------------------------------------------------------------------------
__global__ void gc_degree_kernel(const int* __restrict__ el,
                                 float* __restrict__ deg_in,
                                 float* __restrict__ deg_out) {
    int e = blockIdx.x * blockDim.x + threadIdx.x;
    if (e < N_EDGES) {
        atomicAdd(&deg_in[el[2 * e]], 1.0f);       // src column
        atomicAdd(&deg_out[el[2 * e + 1]], 1.0f);  // dst column
    }
}

// ---------------------------------------------------------------------------
// Kernel 2: normalized gather + scatter-add. One wave32 per (edge | self-loop):
// each lane moves a float4 -> 512B coalesced gather, 4 non-returning
// global_atomic_add_f32 per lane into the L2-resident accumulator.
// ---------------------------------------------------------------------------
__global__ __launch_bounds__(256)
void gc_scatter_kernel(const int* __restrict__ el,
                       const float* __restrict__ x,
                       const float* __restrict__ deg_in,
                       const float* __restrict__ deg_out,
                       float* __restrict__ upd) {
    const int wid  = (int)((blockIdx.x * blockDim.x + threadIdx.x) >> 5);
    const int lane = threadIdx.x & 31;
    if (wid >= N_EDGES + N_NODES) return;

    int src, dst;
    if (wid < N_EDGES) { src = el[2 * wid]; dst = el[2 * wid + 1]; }
    else               { src = dst = wid - N_EDGES; }          // appended self-loop

    const float di = deg_in[src]  + 1.0f;   // +1 for self loop (reference)
    const float dd = deg_out[dst] + 1.0f;
    const float w  = 1.0f / (sqrtf(di * dd) + 1e-10f);

    const float4 xv = *(const float4*)(x + (size_t)src * DIM + lane * 4);
    float* o = upd + (size_t)dst * DIM + lane * 4;
    atomicAdd(o + 0, xv.x * w);
    atomicAdd(o + 1, xv.y * w);
    atomicAdd(o + 2, xv.z * w);
    atomicAdd(o + 3, xv.w * w);
}

// ---------------------------------------------------------------------------
// Kernel 3: out = relu(update @ W^T + b) via V_WMMA_F32_16X16X4_F32.
// Block = 8 waves; block stages one 16x128 band of `update` into padded LDS
// (reads complete before the barrier), each wave owns one 16-col tile of W^T
// and issues 32 fp32 WMMAs (K=128), then writes relu(acc+bias) back in place.
// ---------------------------------------------------------------------------
__global__ __launch_bounds__(256)
void gc_gemm_relu_kernel(const float* __restrict__ Wm,
                         const float* __restrict__ bias,
                         float* __restrict__ out /* in: update, out: result */) {
    __shared__ float ldsA[16 * LDS_STRIDE];

    const int tid      = threadIdx.x;
    const int row_base = blockIdx.x * 16;

    // Stage A band: 16 rows x 128 f32 = 8KB, 2 float4 per thread, coalesced.
    {
        const int r = tid >> 4;
        const int c = (tid & 15) * 8;
        const float4* g = (const float4*)(out + (size_t)(row_base + r) * DIM + c);
        float4 v0 = g[0];
        float4 v1 = g[1];
        *(float4*)&ldsA[r * LDS_STRIDE + c]     = v0;
        *(float4*)&ldsA[r * LDS_STRIDE + c + 4] = v1;
    }
    __syncthreads();

    const int lane  = tid & 31;
    const int wave  = tid >> 5;
    const int ncol  = wave * 16 + (lane & 15);   // output column this lane owns
    const int khalf = (lane >> 4) * 2;           // lanes 16-31 hold K+2,K+3

    const float* ldsrow = &ldsA[(lane & 15) * LDS_STRIDE + khalf];
    const float* wrow   = Wm + (size_t)ncol * DIM + khalf;  // B[k][n] = W[n][k]

    v8f acc = {};
#pragma unroll
    for (int kc = 0; kc < DIM / 4; ++kc) {
        v2f a = *(const v2f*)(ldsrow + kc * 4);  // A[m][k..k+1] (m = lane&15)
        v2f b = *(const v2f*)(wrow + kc * 4);    // B[k..k+1][n]
        acc = __builtin_amdgcn_wmma_f32_16x16x4_f32(
            /*neg_a=*/false, a, /*neg_b=*/false, b,
            /*c_mod=*/(short)0, acc, /*reuse_a=*/false, /*reuse_b=*/false);
    }

    const float bv = bias[ncol];
    float* obase = out + (size_t)row_base * DIM + ncol;
#pragma unroll
    for (int v = 0; v < 8; ++v) {
        const int m = v + (lane >> 4) * 8;       // D VGPR v -> row v (+8 for hi lanes)
        float val = acc[v] + bv;
        obase[(size_t)m * DIM] = val > 0.0f ? val : 0.0f;
    }
}

// ---------------------------------------------------------------------------
extern "C" void kernel_launch(void* const* d_in, const int* in_sizes, int n_in,
                              void* d_out, int out_size, void* d_ws, size_t ws_size,
                              hipStream_t stream) {
    const int*   el = (const int*)d_in[0];    // edge_list [E,2] int32
    const float* x  = (const float*)d_in[1];  // [N,128] f32
    const float* Wm = (const float*)d_in[2];  // [128,128] f32
    const float* b  = (const float*)d_in[3];  // [128] f32
    float* out = (float*)d_out;               // [N,128] f32 (also update scratch)

    float* deg_in  = (float*)d_ws;            // N floats
    float* deg_out = deg_in + N_NODES;        // N floats

    gc_zero_kernel<<<4096, 256, 0, stream>>>(deg_in, out);
    gc_degree_kernel<<<(N_EDGES + 255) / 256, 256, 0, stream>>>(el, deg_in, deg_out);

    const int nmsg = N_EDGES + N_NODES;       // edges + self loops, 1 wave each
    gc_scatter_kernel<<<(nmsg + 7) / 8, 256, 0, stream>>>(el, x, deg_in, deg_out, out);

    gc_gemm_relu_kernel<<<N_NODES / 16, 256, 0, stream>>>(Wm, b, out);
}